// GxEGAT_20478404067769
// MI455X (gfx1250) — compile-verified
//
#include <hip/hip_runtime.h>
#include <hip/hip_bf16.h>

#define NNODES 10000
#define NEDGES 100000
#define NB     64
#define HIDC   128
#define E2TOT  (NEDGES + NNODES)

typedef float v2f __attribute__((ext_vector_type(2)));
typedef float v8f __attribute__((ext_vector_type(8)));

static __device__ __forceinline__ float leaky02(float x){ return x > 0.f ? x : 0.2f * x; }

static __device__ __forceinline__ void atomicMaxF(float* addr, float val){
  int* ai = (int*)addr;
  int old = __float_as_int(*addr);
  while (__int_as_float(old) < val){
    int assumed = old;
    old = atomicCAS(ai, assumed, __float_as_int(val));
    if (old == assumed) break;
  }
}

// ---------------------------------------------------------------- utilities
__global__ void fill_kernel(float* p, float v, long n){
  long i = (long)blockIdx.x * blockDim.x + threadIdx.x;
  if (i < n) p[i] = v;
}

// deg[dst] += 1 ; lsum[dst] += edge_attr  (for self-loop fill='mean')
__global__ void deg_loop_kernel(const int* __restrict__ dst, const float* __restrict__ ea,
                                float* __restrict__ deg, float* __restrict__ lsum, int E){
  int e = blockIdx.x * blockDim.x + threadIdx.x;
  if (e < E){
    int d = dst[e];
    atomicAdd(&deg[d], 1.f);
    atomicAdd(&lsum[d], ea[e]);
  }
}

__global__ void loopattr_kernel(const float* __restrict__ deg, const float* __restrict__ lsum,
                                float* __restrict__ lattr, int N){
  int n = blockIdx.x * blockDim.x + threadIdx.x;
  if (n < N) lattr[n] = lsum[n] / fmaxf(deg[n], 1.f);
}

// h[n,j] = x[n]*vp_w[j] + vp_b[j] + type_emb[node_type[n], j]
__global__ void embed_kernel(const float* __restrict__ x, const int* __restrict__ nt,
                             const float* __restrict__ vpw, const float* __restrict__ vpb,
                             const float* __restrict__ temb, float* __restrict__ h, int N){
  long i = (long)blockIdx.x * blockDim.x + threadIdx.x;
  if (i < (long)N * HIDC){
    int n = (int)(i / HIDC), j = (int)(i % HIDC);
    h[i] = x[n] * vpw[j] + vpb[j] + temb[nt[n] * HIDC + j];
  }
}

// ---------------------------------------------------------------- WMMA GEMM
// C[Nrows x M] = A[Nrows x K] @ W[K x M] + bias[M], all row-major fp32.
// One wave32 computes a 16x64 output strip (4 accumulator tiles along N),
// reusing each 16x4 A fragment across 4 V_WMMA_F32_16X16X4_F32 ops.
// Requires Nrows%16==0, M%64==0, K%4==0 (true for every call here).
// A frag (16x4): lane m<16 -> {A[m][0],A[m][1]}, lane m+16 -> {A[m][2],A[m][3]}
// B frag (4x16): v0: lanes0-15 B[0][n] / lanes16-31 B[2][n]; v1: B[1][n]/B[3][n]
// C/D (16x16):  vgpr r: lanes0-15 row r, lanes16-31 row r+8, col = lane&15
__global__ void gemm_wmma_f32(const float* __restrict__ A, const float* __restrict__ W,
                              const float* __restrict__ bias, float* __restrict__ C,
                              int Nrows, int K, int M){
  const int  lane   = threadIdx.x & 31;
  const long wave   = ((long)blockIdx.x * blockDim.x + threadIdx.x) >> 5;
  const int  strips = M >> 6;                       // strips of 64 columns
  const long total  = (long)(Nrows >> 4) * strips;
  if (wave >= total) return;                        // whole-wave uniform: EXEC all-ones
  const int tm = (int)(wave / strips);
  const int tn = (int)(wave % strips);
  const int r15     = lane & 15;
  const int row     = tm * 16 + r15;
  const int colbase = tn * 64 + r15;
  const int khalf   = (lane >> 4) << 1;             // 0 or 2

  v8f acc0 = {}, acc1 = {}, acc2 = {}, acc3 = {};
  const float* arow = A + (long)row * K + khalf;
  for (int k = 0; k < K; k += 4){
    v2f a = *(const v2f*)(arow + k);                // 8B-aligned
    __builtin_prefetch(arow + k + 64, 0, 0);        // global_prefetch_b8, stream A
    const float* w0 = W + (long)(k + khalf) * M + colbase;
    const float* w1 = w0 + M;
    v2f b;
    b.x = w0[0];  b.y = w1[0];
    acc0 = __builtin_amdgcn_wmma_f32_16x16x4_f32(false, a, false, b, (short)0, acc0, false, false);
    b.x = w0[16]; b.y = w1[16];
    acc1 = __builtin_amdgcn_wmma_f32_16x16x4_f32(false, a, false, b, (short)0, acc1, false, false);
    b.x = w0[32]; b.y = w1[32];
    acc2 = __builtin_amdgcn_wmma_f32_16x16x4_f32(false, a, false, b, (short)0, acc2, false, false);
    b.x = w0[48]; b.y = w1[48];
    acc3 = __builtin_amdgcn_wmma_f32_16x16x4_f32(false, a, false, b, (short)0, acc3, false, false);
  }
  const int rbase = tm * 16 + ((lane >> 4) << 3);
  const float bv0 = bias[colbase],      bv1 = bias[colbase + 16];
  const float bv2 = bias[colbase + 32], bv3 = bias[colbase + 48];
  #pragma unroll
  for (int r = 0; r < 8; ++r){
    float* crow = C + (long)(rbase + r) * M + colbase;
    crow[0]  = acc0[r] + bv0;
    crow[16] = acc1[r] + bv1;
    crow[32] = acc2[r] + bv2;
    crow[48] = acc3[r] + bv3;
  }
}

// ---------------------------------------------------------------- GATv2 edge passes
// One wave32 per (edge, head). out_ch is always 128 = 32 lanes x float4.
__global__ void edge_score_kernel(const float* __restrict__ xl, const float* __restrict__ xr,
                                  const int* __restrict__ src, const int* __restrict__ dst,
                                  const float* __restrict__ ea, const float* __restrict__ lattr,
                                  const float* __restrict__ we, const float* __restrict__ att,
                                  float* __restrict__ score, float* __restrict__ smax,
                                  int E, int E2, int H, int HC){
  const long gw  = ((long)blockIdx.x * blockDim.x + threadIdx.x) >> 5;
  const int lane = threadIdx.x & 31;
  if (gw >= (long)E2 * H) return;
  const int e = (int)(gw / H), h = (int)(gw % H);
  int s, d; float ev;
  if (e < E){ s = src[e]; d = dst[e]; ev = ea[e]; }
  else      { s = e - E;  d = e - E;  ev = lattr[e - E]; }
  const int c0 = lane << 2;                          // one float4 per lane
  const float4 L4 = *(const float4*)(xl  + (long)s * HC + h * HIDC + c0);
  const float4 R4 = *(const float4*)(xr  + (long)d * HC + h * HIDC + c0);
  const float4 W4 = *(const float4*)(we  + h * HIDC + c0);
  const float4 A4 = *(const float4*)(att + h * HIDC + c0);
  float acc = A4.x * leaky02(L4.x + R4.x + ev * W4.x)
            + A4.y * leaky02(L4.y + R4.y + ev * W4.y)
            + A4.z * leaky02(L4.z + R4.z + ev * W4.z)
            + A4.w * leaky02(L4.w + R4.w + ev * W4.w);
  #pragma unroll
  for (int m = 16; m >= 1; m >>= 1) acc += __shfl_xor(acc, m, 32);
  if (lane == 0){
    score[gw] = acc;
    atomicMaxF(&smax[(long)d * H + h], acc);
  }
}

__global__ void edge_exp_kernel(const float* __restrict__ score, const int* __restrict__ dst,
                                const float* __restrict__ smax, float* __restrict__ ex,
                                float* __restrict__ den, int E, int E2, int H){
  long i = (long)blockIdx.x * blockDim.x + threadIdx.x;
  if (i < (long)E2 * H){
    int e = (int)(i / H), h = (int)(i % H);
    int d = (e < E) ? dst[e] : (e - E);
    float v = __expf(score[i] - smax[(long)d * H + h]);
    ex[i] = v;
    atomicAdd(&den[(long)d * H + h], v);
  }
}

__global__ void edge_scatter_kernel(const float* __restrict__ xl, const int* __restrict__ src,
                                    const int* __restrict__ dst, const float* __restrict__ ex,
                                    const float* __restrict__ den, float* __restrict__ out,
                                    int E, int E2, int H, int HC){
  const long gw  = ((long)blockIdx.x * blockDim.x + threadIdx.x) >> 5;
  const int lane = threadIdx.x & 31;
  if (gw >= (long)E2 * H) return;
  const int e = (int)(gw / H), h = (int)(gw % H);
  int s, d;
  if (e < E){ s = src[e]; d = dst[e]; } else { s = d = e - E; }
  const float alpha = ex[gw] / den[(long)d * H + h];
  const int c0 = lane << 2;
  const float4 L4 = *(const float4*)(xl + (long)s * HC + h * HIDC + c0);
  float* po = out + (long)d * HC + h * HIDC + c0;
  atomicAdd(po + 0, alpha * L4.x);
  atomicAdd(po + 1, alpha * L4.y);
  atomicAdd(po + 2, alpha * L4.z);
  atomicAdd(po + 3, alpha * L4.w);
}

__global__ void bias_act_kernel(float* __restrict__ out, const float* __restrict__ bias,
                                long n, int HC){
  long i = (long)blockIdx.x * blockDim.x + threadIdx.x;
  if (i < n) out[i] = leaky02(out[i] + bias[i % HC]);
}

// ---------------------------------------------------------------- pooling + head
__global__ void pool_accum_kernel(const float* __restrict__ h, const int* __restrict__ nt,
                                  const int* __restrict__ batch, float* __restrict__ sums,
                                  float* __restrict__ cnts, int N){
  long i = (long)blockIdx.x * blockDim.x + threadIdx.x;
  if (i < (long)N * HIDC){
    int n = (int)(i / HIDC), j = (int)(i % HIDC);
    if (nt[n] == 0){
      atomicAdd(&sums[batch[n] * HIDC + j], h[i]);
      if (j == 0) atomicAdd(&cnts[batch[n]], 1.f);
    }
  }
}

__global__ void pooled_kernel(const float* __restrict__ sums, const float* __restrict__ cnts,
                              float* __restrict__ pooled){
  int i = blockIdx.x * blockDim.x + threadIdx.x;
  if (i < NB * HIDC){
    int b = i / HIDC;
    pooled[i] = sums[i] / fmaxf(cnts[b], 1.f);
  }
}

// One block (128 threads) per graph: MLP + LayerNorm + final projection.
__global__ void head_kernel(const float* __restrict__ pooled,
                            const float* __restrict__ m1w, const float* __restrict__ m1b,
                            const float* __restrict__ g1,  const float* __restrict__ b1,
                            const float* __restrict__ m2w, const float* __restrict__ m2b,
                            const float* __restrict__ g2,  const float* __restrict__ b2,
                            const float* __restrict__ m3w, const float* __restrict__ m3b,
                            float* __restrict__ out){
  __shared__ float v[HIDC], red[HIDC], h1[HIDC];
  const int b = blockIdx.x, t = threadIdx.x;
  v[t] = pooled[b * HIDC + t];
  __syncthreads();

  float acc = m1b[t];
  for (int k = 0; k < HIDC; ++k) acc += v[k] * m1w[k * HIDC + t];

  red[t] = acc;
  for (int s = 64; s >= 1; s >>= 1){ __syncthreads(); if (t < s) red[t] += red[t + s]; }
  __syncthreads();
  float mu = red[0] * (1.f / HIDC);
  __syncthreads();
  red[t] = (acc - mu) * (acc - mu);
  for (int s = 64; s >= 1; s >>= 1){ __syncthreads(); if (t < s) red[t] += red[t + s]; }
  __syncthreads();
  float var = red[0] * (1.f / HIDC);
  __syncthreads();
  h1[t] = leaky02((acc - mu) * rsqrtf(var + 1e-5f) * g1[t] + b1[t]);
  __syncthreads();

  float acc2 = 0.f;
  if (t < 64){
    acc2 = m2b[t];
    for (int k = 0; k < HIDC; ++k) acc2 += h1[k] * m2w[k * 64 + t];
  }
  red[t] = (t < 64) ? acc2 : 0.f;
  for (int s = 64; s >= 1; s >>= 1){ __syncthreads(); if (t < s) red[t] += red[t + s]; }
  __syncthreads();
  float mu2 = red[0] * (1.f / 64.f);
  __syncthreads();
  float d2 = (t < 64) ? (acc2 - mu2) : 0.f;
  red[t] = d2 * d2;
  for (int s = 64; s >= 1; s >>= 1){ __syncthreads(); if (t < s) red[t] += red[t + s]; }
  __syncthreads();
  float var2 = red[0] * (1.f / 64.f);
  __syncthreads();
  float h2v = 0.f;
  if (t < 64)
    h2v = leaky02((acc2 - mu2) * rsqrtf(var2 + 1e-5f) * g2[t] + b2[t]);
  red[t] = (t < 64) ? h2v * m3w[t] : 0.f;
  for (int s = 64; s >= 1; s >>= 1){ __syncthreads(); if (t < s) red[t] += red[t + s]; }
  __syncthreads();
  if (t == 0) out[b] = red[0] + m3b[0];
}

// ---------------------------------------------------------------- host side
static inline unsigned gblk(long n, int b){ return (unsigned)((n + b - 1) / b); }

struct LayerBufs {
  float *xl, *xr, *score, *ex, *smax, *den, *lattr;
  const int *src, *dst;
  const float *ea;
};

static void run_gat_layer(hipStream_t stream, const LayerBufs& L,
                          const float* hin, int fin, int H,
                          const float* wl, const float* bl,
                          const float* wr, const float* br,
                          const float* we, const float* att,
                          const float* bias, float* hout){
  const int HC = H * HIDC;
  const int E2 = E2TOT;
  // xl = hin@wl + bl ; xr = hin@wr + br  (WMMA, 16x64 strips)
  long waves_g = (long)(NNODES / 16) * (HC / 64);
  long gthr    = waves_g * 32;
  gemm_wmma_f32<<<gblk(gthr, 256), 256, 0, stream>>>(hin, wl, bl, L.xl, NNODES, fin, HC);
  gemm_wmma_f32<<<gblk(gthr, 256), 256, 0, stream>>>(hin, wr, br, L.xr, NNODES, fin, HC);
  // init segment-softmax state + output accumulator
  fill_kernel<<<gblk((long)NNODES * H, 256), 256, 0, stream>>>(L.smax, -1e30f, (long)NNODES * H);
  fill_kernel<<<gblk((long)NNODES * H, 256), 256, 0, stream>>>(L.den, 0.f, (long)NNODES * H);
  fill_kernel<<<gblk((long)NNODES * HC, 256), 256, 0, stream>>>(hout, 0.f, (long)NNODES * HC);
  // attention
  long waves = (long)E2 * H;
  edge_score_kernel<<<gblk(waves * 32, 256), 256, 0, stream>>>(
      L.xl, L.xr, L.src, L.dst, L.ea, L.lattr, we, att, L.score, L.smax, NEDGES, E2, H, HC);
  edge_exp_kernel<<<gblk(waves, 256), 256, 0, stream>>>(
      L.score, L.dst, L.smax, L.ex, L.den, NEDGES, E2, H);
  edge_scatter_kernel<<<gblk(waves * 32, 256), 256, 0, stream>>>(
      L.xl, L.src, L.dst, L.ex, L.den, hout, NEDGES, E2, H, HC);
  bias_act_kernel<<<gblk((long)NNODES * HC, 256), 256, 0, stream>>>(
      hout, bias, (long)NNODES * HC, HC);
}

extern "C" void kernel_launch(void* const* d_in, const int* in_sizes, int n_in,
                              void* d_out, int out_size, void* d_ws, size_t ws_size,
                              hipStream_t stream){
  const float* x          = (const float*)d_in[0];
  const int*   node_type  = (const int*)  d_in[1];
  const int*   edge_index = (const int*)  d_in[2];
  const float* edge_attr  = (const float*)d_in[3];
  const int*   batch      = (const int*)  d_in[4];
  const float* vp_w       = (const float*)d_in[5];
  const float* vp_b       = (const float*)d_in[6];
  const float* type_emb   = (const float*)d_in[7];
  const float* c1_wl = (const float*)d_in[8],  *c1_bl = (const float*)d_in[9];
  const float* c1_wr = (const float*)d_in[10], *c1_br = (const float*)d_in[11];
  const float* c1_we = (const float*)d_in[12], *c1_att = (const float*)d_in[13];
  const float* c1_bias = (const float*)d_in[14];
  const float* c2_wl = (const float*)d_in[15], *c2_bl = (const float*)d_in[16];
  const float* c2_wr = (const float*)d_in[17], *c2_br = (const float*)d_in[18];
  const float* c2_we = (const float*)d_in[19], *c2_att = (const float*)d_in[20];
  const float* c2_bias = (const float*)d_in[21];
  const float* c3_wl = (const float*)d_in[22], *c3_bl = (const float*)d_in[23];
  const float* c3_wr = (const float*)d_in[24], *c3_br = (const float*)d_in[25];
  const float* c3_we = (const float*)d_in[26], *c3_att = (const float*)d_in[27];
  const float* c3_bias = (const float*)d_in[28];
  const float* m1_w = (const float*)d_in[29], *m1_b = (const float*)d_in[30];
  const float* ln1_g = (const float*)d_in[31], *ln1_b = (const float*)d_in[32];
  const float* m2_w = (const float*)d_in[33], *m2_b = (const float*)d_in[34];
  const float* ln2_g = (const float*)d_in[35], *ln2_b = (const float*)d_in[36];
  const float* m3_w = (const float*)d_in[37], *m3_b = (const float*)d_in[38];
  float* out = (float*)d_out;

  const int* src = edge_index;           // edge_index[0, :]
  const int* dst = edge_index + NEDGES;  // edge_index[1, :]

  // workspace carve-up (floats)
  float* ws = (float*)d_ws;
  const size_t NH = (size_t)NNODES * 1024;  // max H*C = 8*128
  float* hbuf  = ws;                 // ping
  float* xl    = ws + NH;
  float* xr    = ws + 2 * NH;
  float* obuf  = ws + 3 * NH;        // pong
  float* score = ws + 4 * NH;        // up to E2*8 = 880,000
  float* ex    = score + 1000000;
  float* smax  = ex + 1000000;       // up to N*8
  float* den   = smax + 100000;
  float* deg   = den + 100000;       // N
  float* lsum  = deg + NNODES;       // N (contiguous with deg)
  float* lattr = lsum + NNODES;      // N
  float* sums  = lattr + NNODES;     // NB*HIDC
  float* cnts  = sums + NB * HIDC;   // NB (contiguous with sums)
  float* pooled = cnts + NB;         // NB*HIDC

  LayerBufs L{xl, xr, score, ex, smax, den, lattr, src, dst, edge_attr};

  // self-loop edge_attr fill = segment-mean over dst
  fill_kernel<<<gblk(2L * NNODES, 256), 256, 0, stream>>>(deg, 0.f, 2L * NNODES);
  deg_loop_kernel<<<gblk(NEDGES, 256), 256, 0, stream>>>(dst, edge_attr, deg, lsum, NEDGES);
  loopattr_kernel<<<gblk(NNODES, 256), 256, 0, stream>>>(deg, lsum, lattr, NNODES);

  // node embedding
  embed_kernel<<<gblk((long)NNODES * HIDC, 256), 256, 0, stream>>>(
      x, node_type, vp_w, vp_b, type_emb, hbuf, NNODES);

  // three GATv2 layers (leaky_relu fused into bias_act)
  run_gat_layer(stream, L, hbuf, HIDC,     8, c1_wl, c1_bl, c1_wr, c1_br, c1_we, c1_att, c1_bias, obuf);
  run_gat_layer(stream, L, obuf, HIDC * 8, 2, c2_wl, c2_bl, c2_wr, c2_br, c2_we, c2_att, c2_bias, hbuf);
  run_gat_layer(stream, L, hbuf, HIDC * 2, 1, c3_wl, c3_bl, c3_wr, c3_br, c3_we, c3_att, c3_bias, obuf);

  // masked mean pool over node_type==0
  fill_kernel<<<gblk((long)NB * HIDC + NB, 256), 256, 0, stream>>>(sums, 0.f, (long)NB * HIDC + NB);
  pool_accum_kernel<<<gblk((long)NNODES * HIDC, 256), 256, 0, stream>>>(
      obuf, node_type, batch, sums, cnts, NNODES);
  pooled_kernel<<<gblk((long)NB * HIDC, 256), 256, 0, stream>>>(sums, cnts, pooled);

  // MLP head -> d_out (64 floats)
  head_kernel<<<NB, HIDC, 0, stream>>>(pooled, m1_w, m1_b, ln1_g, ln1_b,
                                       m2_w, m2_b, ln2_g, ln2_b, m3_w, m3_b, out);
}